// RFCBlock_89318139887948
// MI455X (gfx1250) — compile-verified
//
#include <hip/hip_runtime.h>
#include <math.h>

typedef __attribute__((ext_vector_type(2))) float v2f;
typedef __attribute__((ext_vector_type(8))) float v8f;

constexpr int Bsz = 128, Csz = 1024, HW = 768, INTER = 512, P = 6;
constexpr int ROWS = Bsz * P; // 768 independent (b,p) rows through the MLP chain

// ---------------- Pass A1: partial spatial-attention logits ----------------
// grid = Bsz*8, block=256. Each block: one batch, one 128-channel chunk,
// 3 pixels per thread (i = tid, tid+256, tid+512). Fully coalesced over i.
__global__ __launch_bounds__(256) void attn_partial(const float* __restrict__ x,
                                                    const float* __restrict__ sa_w,
                                                    float* __restrict__ partial) {
  const int tid = threadIdx.x;
  const int b = blockIdx.x >> 3, ch = blockIdx.x & 7;
  const int c0 = ch * 128;
  __shared__ float saw[128];
  if (tid < 128) saw[tid] = sa_w[c0 + tid];
  __syncthreads();
  const float* xb = x + ((size_t)b * Csz + c0) * HW;
  float a0 = 0.f, a1 = 0.f, a2 = 0.f;
#pragma unroll 4
  for (int c = 0; c < 128; ++c) {
    const float wv = saw[c];
    const float* xc = xb + (size_t)c * HW;
    a0 = fmaf(xc[tid], wv, a0);
    a1 = fmaf(xc[tid + 256], wv, a1);
    a2 = fmaf(xc[tid + 512], wv, a2);
  }
  float* pp = partial + ((size_t)b * 8 + ch) * HW;
  pp[tid] = a0; pp[tid + 256] = a1; pp[tid + 512] = a2;
}

// ---------------- Pass A2: sigmoid, normalize, build mw = mnorm*(maps+1) ----
// grid = Bsz, block = 256.
__global__ __launch_bounds__(256) void attn_finalize(const float* __restrict__ partial,
                                                     const float* __restrict__ pm,
                                                     const float* __restrict__ sa_b,
                                                     const float* __restrict__ sa_scale,
                                                     const float* __restrict__ sa_shift,
                                                     float* __restrict__ mw) {
  const int b = blockIdx.x, tid = threadIdx.x;
  __shared__ float red[256];
  __shared__ float sh[8];
  const float sb = sa_b[0], sc = sa_scale[0], ssh = sa_shift[0];
  float fg[3];
#pragma unroll
  for (int j = 0; j < 3; ++j) {
    const int i = tid + 256 * j;
    float lg = 0.f;
#pragma unroll
    for (int ch = 0; ch < 8; ++ch) lg += partial[((size_t)b * 8 + ch) * HW + i];
    lg = fmaf(lg + sb, sc, ssh);
    fg[j] = 1.f / (1.f + expf(-lg));
  }
  red[tid] = fg[0] + fg[1] + fg[2];
  __syncthreads();
  for (int s = 128; s > 0; s >>= 1) { if (tid < s) red[tid] += red[tid + s]; __syncthreads(); }
  if (tid == 0) sh[0] = red[0];
  __syncthreads();
  for (int p = 0; p < P; ++p) {
    float s0 = 0.f;
#pragma unroll
    for (int j = 0; j < 3; ++j) s0 += pm[((size_t)b * P + p) * HW + tid + 256 * j];
    red[tid] = s0;
    __syncthreads();
    for (int s = 128; s > 0; s >>= 1) { if (tid < s) red[tid] += red[tid + s]; __syncthreads(); }
    if (tid == 0) sh[1 + p] = red[0];
    __syncthreads();
  }
  const float invT = 1.f / sh[0];
#pragma unroll
  for (int p = 0; p < P; ++p) {
    const float invm = 1.f / sh[1 + p];
#pragma unroll
    for (int j = 0; j < 3; ++j) {
      const int i = tid + 256 * j;
      const size_t idx = ((size_t)b * P + p) * HW + i;
      mw[idx] = pm[idx] * invm * (1.f + fg[j] * invT);
    }
  }
}

// ---------------- Pass B: part pooling  t[(b*6+p)][c] = sum_i mw[p,i]*x[c,i] --
// grid = Bsz*(Csz/8). One wave per channel, lanes stride pixels (coalesced).
__global__ __launch_bounds__(256) void pool_kernel(const float* __restrict__ x,
                                                   const float* __restrict__ mw,
                                                   float* __restrict__ t) {
  const int blk = blockIdx.x;
  const int b = blk >> 7, c0 = (blk & 127) * 8;
  __shared__ float m[P * HW];
  for (int j = threadIdx.x; j < P * HW; j += 256) m[j] = mw[(size_t)b * P * HW + j];
  __syncthreads();
  const int w = threadIdx.x >> 5, lane = threadIdx.x & 31;
  const int c = c0 + w;
  const float* xc = x + ((size_t)b * Csz + c) * HW;
  float a0 = 0, a1 = 0, a2 = 0, a3 = 0, a4 = 0, a5 = 0;
#pragma unroll 4
  for (int k = 0; k < HW / 32; ++k) {
    const int i = lane + k * 32;
    const float xv = xc[i];
    a0 = fmaf(m[i], xv, a0);          a1 = fmaf(m[HW + i], xv, a1);
    a2 = fmaf(m[2 * HW + i], xv, a2); a3 = fmaf(m[3 * HW + i], xv, a3);
    a4 = fmaf(m[4 * HW + i], xv, a4); a5 = fmaf(m[5 * HW + i], xv, a5);
  }
#pragma unroll
  for (int off = 16; off > 0; off >>= 1) {
    a0 += __shfl_xor(a0, off); a1 += __shfl_xor(a1, off); a2 += __shfl_xor(a2, off);
    a3 += __shfl_xor(a3, off); a4 += __shfl_xor(a4, off); a5 += __shfl_xor(a5, off);
  }
  if (lane == 0) {
    float* tp = t + (size_t)b * P * Csz + c;
    tp[0] = a0; tp[Csz] = a1; tp[2 * Csz] = a2;
    tp[3 * Csz] = a3; tp[4 * Csz] = a4; tp[5 * Csz] = a5;
  }
}

// ---------------- WMMA f32 GEMM: C[M,N] = act(A@B + bias)*scale + res --------
// BL=1: B stored [N][K] row-major (i.e. use B^T), BL=0: B stored [K][N].
template <int BL, int RELU, int RES, int SCALE>
__global__ __launch_bounds__(256) void gemm_wmma(const float* __restrict__ A,
                                                 const float* __restrict__ Bm,
                                                 const float* __restrict__ bias,
                                                 const float* __restrict__ res,
                                                 const float* __restrict__ scalev,
                                                 float* __restrict__ C,
                                                 int M, int N, int K) {
  constexpr int BM = 128, BN = 64, BK = 32, LDT = BK + 4; // stride 36: conflict-free v2f reads
  __shared__ float As[BM * LDT];
  __shared__ float Bs[BN * LDT];
  const int tid = threadIdx.x;
  const int m0 = blockIdx.y * BM, n0 = blockIdx.x * BN;
  const int w = tid >> 5, lane = tid & 31;
  const int wm = (w >> 1) * 32, wn = (w & 1) * 32; // 4x2 waves, each 32x32
  const int half = lane >> 4, lm = lane & 15;
  v8f acc[2][2] = {};
  for (int kt = 0; kt < K; kt += BK) {
#pragma unroll
    for (int j = 0; j < 4; ++j) { // A tile 128x32
      const int idx = tid + 256 * j, r = idx >> 3, c4 = (idx & 7) * 4;
      *reinterpret_cast<float4*>(&As[r * LDT + c4]) =
          *reinterpret_cast<const float4*>(&A[(size_t)(m0 + r) * K + kt + c4]);
    }
    if (BL == 1) {
#pragma unroll
      for (int j = 0; j < 2; ++j) { // B tile as [n][k] straight copy
        const int idx = tid + 256 * j, r = idx >> 3, c4 = (idx & 7) * 4;
        *reinterpret_cast<float4*>(&Bs[r * LDT + c4]) =
            *reinterpret_cast<const float4*>(&Bm[(size_t)(n0 + r) * K + kt + c4]);
      }
    } else {
#pragma unroll
      for (int j = 0; j < 2; ++j) { // B [K][N] -> transpose into [n][k]
        const int idx = tid + 256 * j, kr = idx >> 4, nc = (idx & 15) * 4;
        const float4 v = *reinterpret_cast<const float4*>(&Bm[(size_t)(kt + kr) * N + n0 + nc]);
        Bs[(nc + 0) * LDT + kr] = v.x; Bs[(nc + 1) * LDT + kr] = v.y;
        Bs[(nc + 2) * LDT + kr] = v.z; Bs[(nc + 3) * LDT + kr] = v.w;
      }
    }
    __syncthreads();
#pragma unroll
    for (int kk = 0; kk < BK / 4; ++kk) {
      const int kc = kk * 4 + half * 2; // lanes 0-15: K=k,k+1 ; lanes 16-31: K=k+2,k+3
      const v2f a0 = *reinterpret_cast<const v2f*>(&As[(wm + lm) * LDT + kc]);
      const v2f a1 = *reinterpret_cast<const v2f*>(&As[(wm + 16 + lm) * LDT + kc]);
      const v2f b0 = *reinterpret_cast<const v2f*>(&Bs[(wn + lm) * LDT + kc]);
      const v2f b1 = *reinterpret_cast<const v2f*>(&Bs[(wn + 16 + lm) * LDT + kc]);
      acc[0][0] = __builtin_amdgcn_wmma_f32_16x16x4_f32(false, a0, false, b0, (short)0, acc[0][0], false, false);
      acc[0][1] = __builtin_amdgcn_wmma_f32_16x16x4_f32(false, a0, false, b1, (short)0, acc[0][1], false, false);
      acc[1][0] = __builtin_amdgcn_wmma_f32_16x16x4_f32(false, a1, false, b0, (short)0, acc[1][0], false, false);
      acc[1][1] = __builtin_amdgcn_wmma_f32_16x16x4_f32(false, a1, false, b1, (short)0, acc[1][1], false, false);
    }
    __syncthreads();
  }
#pragma unroll
  for (int ms = 0; ms < 2; ++ms)
#pragma unroll
    for (int ns = 0; ns < 2; ++ns) {
      const int col = n0 + wn + ns * 16 + lm;
      const int rb = m0 + wm + ms * 16 + half * 8; // C/D layout: VGPR r -> row r (+8 for hi lanes)
      const float bv = bias ? bias[col] : 0.f;
      const float sv = SCALE ? scalev[col] : 1.f;
#pragma unroll
      for (int r = 0; r < 8; ++r) {
        float v = acc[ms][ns][r] + bv;
        if (RELU) v = v > 0.f ? v : 0.f;
        if (SCALE) v *= sv;
        if (RES) v += res[(size_t)(rb + r) * N + col];
        C[(size_t)(rb + r) * N + col] = v;
      }
    }
}

// ---------------- Pass D: rank-6 scatter + extend affine + residual ----------
__global__ __launch_bounds__(256) void out_kernel(const float* __restrict__ x,
                                                  const float* __restrict__ pm,
                                                  const float* __restrict__ zs,
                                                  const float* __restrict__ eb,
                                                  const float* __restrict__ esc,
                                                  const float* __restrict__ esh,
                                                  float* __restrict__ out) {
  const int blk = blockIdx.x;
  const int b = blk >> 7, c0 = (blk & 127) * 8;
  __shared__ float m[P * HW];
  for (int j = threadIdx.x; j < P * HW; j += 256) m[j] = pm[(size_t)b * P * HW + j];
  __syncthreads();
  const int w = threadIdx.x >> 5, lane = threadIdx.x & 31;
  const int c = c0 + w;
  const float* zp = zs + (size_t)b * P * Csz + c;
  const float z0 = zp[0], z1 = zp[Csz], z2 = zp[2 * Csz];
  const float z3 = zp[3 * Csz], z4 = zp[4 * Csz], z5 = zp[5 * Csz];
  const float ebc = fmaf(eb[c], esc[c], esh[c]);
  const float* xc = x + ((size_t)b * Csz + c) * HW;
  float* oc = out + ((size_t)b * Csz + c) * HW;
#pragma unroll 4
  for (int k = 0; k < HW / 32; ++k) {
    const int i = lane + k * 32;
    float v = xc[i] + ebc;
    v = fmaf(m[i], z0, v);          v = fmaf(m[HW + i], z1, v);
    v = fmaf(m[2 * HW + i], z2, v); v = fmaf(m[3 * HW + i], z3, v);
    v = fmaf(m[4 * HW + i], z4, v); v = fmaf(m[5 * HW + i], z5, v);
    oc[i] = v;
  }
}

extern "C" void kernel_launch(void* const* d_in, const int* in_sizes, int n_in,
                              void* d_out, int out_size, void* d_ws, size_t ws_size,
                              hipStream_t stream) {
  const float* x        = (const float*)d_in[0];
  const float* pm       = (const float*)d_in[1];
  const float* sa_w     = (const float*)d_in[2];
  const float* sa_b     = (const float*)d_in[3];
  const float* sa_scale = (const float*)d_in[4];
  const float* sa_shift = (const float*)d_in[5];
  const float* reduce_w = (const float*)d_in[6];
  const float* reduce_b = (const float*)d_in[7];
  const float* w1       = (const float*)d_in[8];
  const float* b1       = (const float*)d_in[9];
  const float* w2       = (const float*)d_in[10];
  const float* b2       = (const float*)d_in[11];
  const float* ext_w    = (const float*)d_in[12];
  const float* ext_b    = (const float*)d_in[13];
  const float* ext_scale= (const float*)d_in[14];
  const float* ext_shift= (const float*)d_in[15];

  float* ws      = (float*)d_ws;
  float* partial = ws;                       // 128*8*768
  float* mw      = partial + Bsz * 8 * HW;   // 128*6*768
  float* t       = mw + Bsz * P * HW;        // 768*1024
  float* pooled  = t + ROWS * Csz;           // 768*512
  float* hidden  = pooled + ROWS * INTER;    // 768*512
  float* yb      = hidden + ROWS * INTER;    // 768*512
  float* zsb     = yb + ROWS * INTER;        // 768*1024  (~16.5 MB total)

  attn_partial<<<Bsz * 8, 256, 0, stream>>>(x, sa_w, partial);
  attn_finalize<<<Bsz, 256, 0, stream>>>(partial, pm, sa_b, sa_scale, sa_shift, mw);
  pool_kernel<<<Bsz * (Csz / 8), 256, 0, stream>>>(x, mw, t);
  // pooled = t @ reduce_w^T + reduce_b        (B is [N=512][K=1024] row-major)
  gemm_wmma<1, 0, 0, 0><<<dim3(INTER / 64, ROWS / 128), 256, 0, stream>>>(
      t, reduce_w, reduce_b, nullptr, nullptr, pooled, ROWS, INTER, Csz);
  // hidden = relu(pooled @ w1 + b1)           (B is [K][N])
  gemm_wmma<0, 1, 0, 0><<<dim3(INTER / 64, ROWS / 128), 256, 0, stream>>>(
      pooled, w1, b1, nullptr, nullptr, hidden, ROWS, INTER, INTER);
  // y = pooled + hidden @ w2 + b2
  gemm_wmma<0, 0, 1, 0><<<dim3(INTER / 64, ROWS / 128), 256, 0, stream>>>(
      hidden, w2, b2, pooled, nullptr, yb, ROWS, INTER, INTER);
  // zs = (y @ ext_w^T) * ext_scale            (B is [N=1024][K=512] row-major)
  gemm_wmma<1, 0, 0, 1><<<dim3(Csz / 64, ROWS / 128), 256, 0, stream>>>(
      yb, ext_w, nullptr, nullptr, ext_scale, zsb, ROWS, Csz, INTER);
  out_kernel<<<Bsz * (Csz / 8), 256, 0, stream>>>(x, pm, zsb, ext_b, ext_scale, ext_shift,
                                                  (float*)d_out);
}